// MaskPowerNorm_2413771621060
// MI455X (gfx1250) — compile-verified
//
#include <hip/hip_runtime.h>
#include <stdint.h>

#define EPS    1e-5f
#define T_DIM  2048
#define B_DIM  16
#define C_DIM  1024
#define NTOK   (T_DIM * B_DIM)   // 32768 tokens

typedef float f32x4 __attribute__((ext_vector_type(4)));

// ---- workspace layout (in floats) -------------------------------------
//  [0, 32768)                  s_tok[n] = rsqrt(mean(x^2)+eps) per token
//  [32768, 32768+1024)         var_sum[c] (masked sum of xg^2 over tokens)
//  [33792]                     count (number of valid tokens, as float)
//  [34048, 34048+256*1024)     per-block partial channel sums (256 blocks)
static constexpr int WS_STOK = 0;
static constexpr int WS_VAR  = NTOK;           // 32768
static constexpr int WS_CNT  = NTOK + C_DIM;   // 33792
static constexpr int WS_PART = 34048;          // 16-float aligned

// ======================================================================
// Kernel 0: count valid tokens from pad_mask [B, T] (bool, 1 byte/elt).
// valid = (pad_mask == 0). One block.
// ======================================================================
__global__ __launch_bounds__(256) void k_count(const unsigned char* __restrict__ pm,
                                               float* __restrict__ ws) {
    __shared__ int red[256];
    const int tid = threadIdx.x;
    int local = 0;
    const uint4* p4 = (const uint4*)pm;        // 32768 bytes = 2048 uint4
    for (int i = tid; i < NTOK / 16; i += 256) {
        uint4 v = p4[i];
        // bytes are 0/1, so popcount counts padding bytes
        int pad = __popc(v.x) + __popc(v.y) + __popc(v.z) + __popc(v.w);
        local += 16 - pad;
    }
    red[tid] = local;
    __syncthreads();
    for (int s = 128; s > 0; s >>= 1) {
        if (tid < s) red[tid] += red[tid + s];
        __syncthreads();
    }
    if (tid == 0) ws[WS_CNT] = (float)red[0];
}

// ======================================================================
// Kernel 1: per-token inverse RMS + masked per-channel partial sums.
// One wave (32 lanes) per token; lane l owns channels {k*128 + 4l .. +3}.
// 256 blocks x 8 waves, 16 tokens per wave.
// ======================================================================
__global__ __launch_bounds__(256) void k_stats(const float* __restrict__ x,
                                               const unsigned char* __restrict__ pm,
                                               float* __restrict__ ws) {
    __shared__ __align__(16) float red[8][C_DIM];   // 32 KB
    const int tid  = threadIdx.x;
    const int lane = tid & 31;
    const int widx = tid >> 5;
    const int wave_global = blockIdx.x * 8 + widx;  // 0..2047
    const int base_ch = lane * 4;
    float* stok = ws + WS_STOK;

    f32x4 acc[8];
#pragma unroll
    for (int k = 0; k < 8; ++k) acc[k] = (f32x4)0.0f;

    for (int i = 0; i < 16; ++i) {
        const int n = wave_global * 16 + i;
        const float* xp = x + (size_t)n * C_DIM;
        // gfx1250 global_prefetch_b8: pull next token's line while we reduce
        __builtin_prefetch(xp + C_DIM + base_ch, 0, 1);

        f32x4 v[8];
        float ss = 0.0f;
#pragma unroll
        for (int k = 0; k < 8; ++k) {
            v[k] = *(const f32x4*)(xp + k * 128 + base_ch);   // global_load_b128
            ss = fmaf(v[k].x, v[k].x, ss);
            ss = fmaf(v[k].y, v[k].y, ss);
            ss = fmaf(v[k].z, v[k].z, ss);
            ss = fmaf(v[k].w, v[k].w, ss);
        }
        // wave32 butterfly reduction (5 steps, warpSize==32 on gfx1250)
#pragma unroll
        for (int m = 16; m >= 1; m >>= 1) ss += __shfl_xor(ss, m, 32);

        const float m2 = ss * (1.0f / C_DIM);
        const float r  = rsqrtf(m2 + EPS);
        if (lane == 0) stok[n] = r;

        // masked accumulate of xg^2 = x^2 * r^2
        const int t = n >> 4, b = n & 15;                    // x is [T,B,C]
        const float mscale = (pm[b * T_DIM + t] == 0) ? (r * r) : 0.0f;
#pragma unroll
        for (int k = 0; k < 8; ++k) {
            acc[k].x = fmaf(v[k].x * mscale, v[k].x, acc[k].x);
            acc[k].y = fmaf(v[k].y * mscale, v[k].y, acc[k].y);
            acc[k].z = fmaf(v[k].z * mscale, v[k].z, acc[k].z);
            acc[k].w = fmaf(v[k].w * mscale, v[k].w, acc[k].w);
        }
    }

    // cross-wave reduction in LDS, then one partial vector per block
#pragma unroll
    for (int k = 0; k < 8; ++k)
        *(f32x4*)&red[widx][k * 128 + base_ch] = acc[k];
    __syncthreads();

    float* part = ws + WS_PART + (size_t)blockIdx.x * C_DIM;
    for (int j = 0; j < 4; ++j) {
        const int c = tid + 256 * j;
        float s = 0.0f;
#pragma unroll
        for (int wv = 0; wv < 8; ++wv) s += red[wv][c];
        part[c] = s;
    }
}

// ======================================================================
// Kernel 2: fold the 256 per-block partials into var_sum[C].
// 4 blocks x 256 threads, one channel per thread, coalesced.
// ======================================================================
__global__ __launch_bounds__(256) void k_reduce(float* __restrict__ ws) {
    const int c = blockIdx.x * 256 + threadIdx.x;   // 0..1023
    const float* part = ws + WS_PART;
    float s = 0.0f;
    for (int b = 0; b < 256; ++b) s += part[(size_t)b * C_DIM + c];
    ws[WS_VAR + c] = s;
}

// ======================================================================
// Kernel 3: apply. out = weight * (x * s_tok) * rsqrt(var + eps) + bias.
// Per-wave double-buffered async global->LDS streaming of x (ASYNCcnt),
// non-temporal b128 stores of out. 1024 blocks x 128 threads (4 waves),
// 8 tokens per wave.
// ======================================================================
__global__ __launch_bounds__(128) void k_apply(const float* __restrict__ x,
                                               const float* __restrict__ weight,
                                               const float* __restrict__ bias,
                                               const float* __restrict__ ws,
                                               float* __restrict__ out) {
    __shared__ __align__(16) float sA[C_DIM];            // weight * inv_std
    __shared__ __align__(16) float sB[C_DIM];            // bias
    __shared__ __align__(16) float xbuf[4][2][C_DIM];    // 32 KB double buffers

    const int tid = threadIdx.x;
    const float invCnt = 1.0f / ws[WS_CNT];
    for (int j = 0; j < 8; ++j) {
        const int c = tid + 128 * j;
        sA[c] = weight[c] * rsqrtf(ws[WS_VAR + c] * invCnt + EPS);
        sB[c] = bias[c];
    }
    __syncthreads();

    const int lane = tid & 31;
    const int widx = tid >> 5;
    const int wave_global = blockIdx.x * 4 + widx;   // 0..4095
    const int base_ch = lane * 4;
    const float* stok = ws + WS_STOK;

    // LDS byte addresses of this lane's slice in each buffer
    const uint32_t lds0 = (uint32_t)(uintptr_t)&xbuf[widx][0][base_ch];
    const uint32_t lds1 = (uint32_t)(uintptr_t)&xbuf[widx][1][base_ch];

    const int NT = 8;
    const int n0 = wave_global * NT;

    // prologue: async-load token 0 into buffer 0 (8 x b128 per lane)
    {
        const float* xp = x + (size_t)n0 * C_DIM + base_ch;
#pragma unroll
        for (int k = 0; k < 8; ++k)
            asm volatile("global_load_async_to_lds_b128 %0, %1, off"
                         :: "v"(lds0 + (uint32_t)(k * 512)), "v"(xp + k * 128)
                         : "memory");
    }

    for (int i = 0; i < NT; ++i) {
        const int n = n0 + i;
        if (i + 1 < NT) {
            const uint32_t dst = ((i + 1) & 1) ? lds1 : lds0;
            const float* xp = x + (size_t)(n + 1) * C_DIM + base_ch;
#pragma unroll
            for (int k = 0; k < 8; ++k)
                asm volatile("global_load_async_to_lds_b128 %0, %1, off"
                             :: "v"(dst + (uint32_t)(k * 512)), "v"(xp + k * 128)
                             : "memory");
            // the 8 just-issued may stay in flight; batch i must be complete
            asm volatile("s_wait_asynccnt 8" ::: "memory");
        } else {
            asm volatile("s_wait_asynccnt 0" ::: "memory");
        }

        const float s = stok[n];                 // wave-uniform
        const float* lbuf = xbuf[widx][i & 1];
        float* op = out + (size_t)n * C_DIM + base_ch;
#pragma unroll
        for (int k = 0; k < 8; ++k) {
            f32x4 v  = *(const f32x4*)(lbuf + k * 128 + base_ch);  // ds_load_b128
            f32x4 a  = *(const f32x4*)(&sA[k * 128 + base_ch]);
            f32x4 bb = *(const f32x4*)(&sB[k * 128 + base_ch]);
            f32x4 o;
            o.x = fmaf(a.x, v.x * s, bb.x);
            o.y = fmaf(a.y, v.y * s, bb.y);
            o.z = fmaf(a.z, v.z * s, bb.z);
            o.w = fmaf(a.w, v.w * s, bb.w);
            __builtin_nontemporal_store(o, (f32x4*)(op + k * 128));  // NT b128
        }
    }
}

// ======================================================================
extern "C" void kernel_launch(void* const* d_in, const int* in_sizes, int n_in,
                              void* d_out, int out_size, void* d_ws, size_t ws_size,
                              hipStream_t stream) {
    (void)in_sizes; (void)n_in; (void)out_size; (void)ws_size;
    const float*         x      = (const float*)d_in[0];
    const float*         weight = (const float*)d_in[1];
    const float*         bias   = (const float*)d_in[2];
    const unsigned char* pmask  = (const unsigned char*)d_in[3]; // bool [B,T]
    float* out = (float*)d_out;
    float* ws  = (float*)d_ws;

    k_count <<<1,    256, 0, stream>>>(pmask, ws);
    k_stats <<<256,  256, 0, stream>>>(x, pmask, ws);
    k_reduce<<<4,    256, 0, stream>>>(ws);
    k_apply <<<1024, 128, 0, stream>>>(x, weight, bias, ws, out);
}